// GraNNy_ViPeR_35820027249494
// MI455X (gfx1250) — compile-verified
//
#include <hip/hip_runtime.h>
#include <hip/hip_bf16.h>

// GNN pipeline for MI455X (gfx1250, wave32).
// - fp32 WMMA 16x16x4 for the 128x128 SAGE GEMMs (precision-exact vs fp32 ref)
// - A-tiles staged to LDS via async global->LDS DMA (ASYNCcnt path) when available
// - L2-resident float atomics for all segment_sums (dest arrays << 192MB L2)
// - sage3 projected-before-aggregate (linear commutes with mean)
// Edge index assumed int32 per harness contract ("integer -> const int*").

typedef float v2f __attribute__((ext_vector_type(2)));
typedef float v8f __attribute__((ext_vector_type(8)));

#if defined(__gfx1250__) && __has_builtin(__builtin_amdgcn_global_load_async_to_lds_b128) && \
    __has_builtin(__builtin_amdgcn_s_wait_asynccnt)
#define ASYNC_LDS 1
// builtin signature (from clang diagnostic): (AS1 v4i*, AS3 v4i*, int offset, int cpol)
typedef int v4i_t __attribute__((vector_size(16)));
typedef __attribute__((address_space(1))) v4i_t* as1_v4i;
typedef __attribute__((address_space(3))) v4i_t* as3_v4i;
#else
#define ASYNC_LDS 0
#endif

#define DIV_UP(a, b) (((a) + (b) - 1) / (b))

__device__ __forceinline__ float sigmoidf_(float v) { return 1.0f / (1.0f + __expf(-v)); }

// ---------- degree / norm ----------
__global__ __launch_bounds__(256) void k_deg(const int* __restrict__ dst, float* __restrict__ deg, int nE) {
    int e = blockIdx.x * 256 + threadIdx.x;
    if (e < nE) atomicAdd(&deg[dst[e]], 1.0f);
}

__global__ __launch_bounds__(256) void k_dn(const float* __restrict__ deg, float* __restrict__ invc,
                                            float* __restrict__ dis, int nN) {
    int i = blockIdx.x * 256 + threadIdx.x;
    if (i < nN) {
        float d = deg[i];
        invc[i] = 1.0f / fmaxf(d, 1.0f);
        dis[i]  = (d > 0.0f) ? rsqrtf(d) : 0.0f;
    }
}

__global__ __launch_bounds__(256) void k_norm(const int* __restrict__ src, const int* __restrict__ dst,
                                              const float* __restrict__ dis, float* __restrict__ nrm, int nE) {
    int e = blockIdx.x * 256 + threadIdx.x;
    if (e < nE) nrm[e] = dis[src[e]] * dis[dst[e]];
}

// ---------- edge scatter-add (thread per edge, C small) ----------
template <int C>
__global__ __launch_bounds__(256) void k_scatter(const int* __restrict__ src, const int* __restrict__ dst,
                                                 const float* __restrict__ w, const float* __restrict__ in,
                                                 float* __restrict__ out, int nE) {
    int e = blockIdx.x * 256 + threadIdx.x;
    if (e >= nE) return;
    int s = src[e], d = dst[e];
    float f = w ? w[e] : 1.0f;
    const float* ip = in + (size_t)s * C;
    float* op = out + (size_t)d * C;
#pragma unroll
    for (int c = 0; c < C; ++c) atomicAdd(&op[c], ip[c] * f);
}

// ---------- edge scatter-add, 128 features, wave per edge ----------
__global__ __launch_bounds__(256) void k_scatter128(const int* __restrict__ src, const int* __restrict__ dst,
                                                    const float* __restrict__ in, float* __restrict__ out, int nE) {
    int e = (blockIdx.x << 3) + (threadIdx.x >> 5);
    if (e >= nE) return;
    int lane = threadIdx.x & 31;
    if (e + 8192 < nE) __builtin_prefetch(&src[e + 8192], 0, 1);  // global_prefetch_b8
    int s = src[e], d = dst[e];
    float4 v = ((const float4*)(in + (size_t)s * 128))[lane];
    float* op = out + (size_t)d * 128 + lane * 4;
    atomicAdd(op + 0, v.x);
    atomicAdd(op + 1, v.y);
    atomicAdd(op + 2, v.z);
    atomicAdd(op + 3, v.w);
}

// ---------- SAGE1: out = sigmoid(mean(agg11)@wl.T + x@wr.T + b), 11 -> 128 ----------
__global__ __launch_bounds__(256) void k_sage1(const float* __restrict__ agg, const float* __restrict__ x,
                                               const float* __restrict__ invc, const float* __restrict__ wl,
                                               const float* __restrict__ wr, const float* __restrict__ b,
                                               float* __restrict__ out, int nN) {
    int idx = blockIdx.x * 256 + threadIdx.x;
    if (idx >= nN * 128) return;
    int i = idx >> 7, o = idx & 127;
    float ic = invc[i];
    const float* ar = agg + (size_t)i * 11;
    const float* xr = x + (size_t)i * 11;
    const float* wlr = wl + o * 11;
    const float* wrr = wr + o * 11;
    float acc = b[o];
#pragma unroll
    for (int c = 0; c < 11; ++c) acc += ar[c] * ic * wlr[c] + xr[c] * wrr[c];
    out[(size_t)i * 128 + o] = sigmoidf_(acc);
}

// ---------- SAGE2: WMMA f32 16x16x4. out = sigmoid((agg@wl.T)*invc + x@wr.T + b) ----------
// Block: 256 thr = 8 waves; block handles 16 nodes, wave w handles out cols [16w,16w+16).
// Raw A tiles (agg, x) staged in LDS (async DMA when available); invc folded into the
// epilogue so the staged tiles are plain byte copies. In-place out==agg is safe: all
// global reads of agg happen before the barrier and each block writes only its own rows.
__global__ __launch_bounds__(256) void k_sage2_wmma(const float* __restrict__ agg, const float* __restrict__ xin,
                                                    const float* __restrict__ invc, const float* __restrict__ wl,
                                                    const float* __restrict__ wr, const float* __restrict__ bias,
                                                    float* __restrict__ out, int nN) {
    __shared__ float sA[16 * 128];  // raw agg tile
    __shared__ float sX[16 * 128];  // raw x tile
    int m0 = blockIdx.x << 4;

#if ASYNC_LDS
    if (m0 + 16 <= nN) {
        // global -> LDS async DMA, 16B per op, 2 ops per thread per array
        for (int idx = threadIdx.x; idx < 16 * 32; idx += 256) {
            int r = idx >> 5, c4 = (idx & 31) << 2;
            const float* ga = agg + (size_t)(m0 + r) * 128 + c4;
            const float* gx = xin + (size_t)(m0 + r) * 128 + c4;
            __builtin_amdgcn_global_load_async_to_lds_b128(
                (as1_v4i)(const __attribute__((address_space(1))) void*)ga,
                (as3_v4i)(__attribute__((address_space(3))) void*)(sA + r * 128 + c4), 0, 0);
            __builtin_amdgcn_global_load_async_to_lds_b128(
                (as1_v4i)(const __attribute__((address_space(1))) void*)gx,
                (as3_v4i)(__attribute__((address_space(3))) void*)(sX + r * 128 + c4), 0, 0);
        }
        __builtin_amdgcn_s_wait_asynccnt(0);
    } else
#endif
    {
        for (int idx = threadIdx.x; idx < 16 * 32; idx += 256) {
            int r = idx >> 5, c4 = (idx & 31) << 2;
            int gr = m0 + r;
            float4 va = make_float4(0.f, 0.f, 0.f, 0.f), vx = va;
            if (gr < nN) {
                va = *(const float4*)(agg + (size_t)gr * 128 + c4);
                vx = *(const float4*)(xin + (size_t)gr * 128 + c4);
            }
            *(float4*)(sA + r * 128 + c4) = va;
            *(float4*)(sX + r * 128 + c4) = vx;
        }
    }
    __syncthreads();

    int wave = threadIdx.x >> 5;
    int lane = threadIdx.x & 31;
    int l15 = lane & 15;
    int hi = lane >> 4;    // 0 or 1
    int khal = hi << 1;    // K sub-offset: 0 or 2
    int n = (wave << 4) + l15;

    const float* wlr = wl + (size_t)n * 128;  // B[k][n] = wl[n][k]
    const float* wrr = wr + (size_t)n * 128;
    const float* aA = sA + l15 * 128;         // A row m = l15
    const float* aX = sX + l15 * 128;

    v8f c1 = {};  // agg @ wl.T
    v8f c2 = {};  // x   @ wr.T

#pragma unroll 8
    for (int k = 0; k < 128; k += 4) {
        v2f a, b;
        a.x = aA[k + khal]; a.y = aA[k + khal + 1];
        b.x = wlr[k + khal]; b.y = wlr[k + khal + 1];
        c1 = __builtin_amdgcn_wmma_f32_16x16x4_f32(false, a, false, b, (short)0, c1, false, false);
    }
#pragma unroll 8
    for (int k = 0; k < 128; k += 4) {
        v2f a, b;
        a.x = aX[k + khal]; a.y = aX[k + khal + 1];
        b.x = wrr[k + khal]; b.y = wrr[k + khal + 1];
        c2 = __builtin_amdgcn_wmma_f32_16x16x4_f32(false, a, false, b, (short)0, c2, false, false);
    }

    float bn = bias[n];
#pragma unroll
    for (int r = 0; r < 8; ++r) {
        int row = m0 + r + (hi << 3);
        if (row < nN) {
            float v = c1[r] * invc[row] + c2[r] + bn;
            out[(size_t)row * 128 + n] = sigmoidf_(v);
        }
    }
}

// ---------- SAGE3 projection: yl = x1b . wl3, yr = x1b . wr3 (wave per node) ----------
__global__ __launch_bounds__(256) void k_sage3_proj(const float* __restrict__ xin, const float* __restrict__ wl,
                                                    const float* __restrict__ wr, float* __restrict__ yl,
                                                    float* __restrict__ yr, int nN) {
    int node = (blockIdx.x << 3) + (threadIdx.x >> 5);
    int lane = threadIdx.x & 31;
    float sl = 0.f, sr = 0.f;
    if (node < nN) {
        float4 v = ((const float4*)(xin + (size_t)node * 128))[lane];
        float4 a = ((const float4*)wl)[lane];
        float4 b = ((const float4*)wr)[lane];
        sl = v.x * a.x + v.y * a.y + v.z * a.z + v.w * a.w;
        sr = v.x * b.x + v.y * b.y + v.z * b.z + v.w * b.w;
    }
#pragma unroll
    for (int off = 16; off > 0; off >>= 1) {
        sl += __shfl_xor(sl, off, 32);
        sr += __shfl_xor(sr, off, 32);
    }
    if (node < nN && lane == 0) { yl[node] = sl; yr[node] = sr; }
}

__global__ __launch_bounds__(256) void k_sage3_fin(const float* __restrict__ aggy, const float* __restrict__ invc,
                                                   const float* __restrict__ yr, const float* __restrict__ b3,
                                                   float* __restrict__ out, int nN) {
    int i = blockIdx.x * 256 + threadIdx.x;
    if (i < nN) out[i] = fmaxf(aggy[i] * invc[i] + yr[i] + b3[0], 0.0f);
}

// ---------- TAG per-hop linear accumulate: out (+)= h @ w_k.T ----------
template <int CIN, int COUT>
__global__ __launch_bounds__(256) void k_tagacc(const float* __restrict__ h, const float* __restrict__ w,
                                                int wstride, const float* __restrict__ b, float* __restrict__ out,
                                                int nN, int init) {
    int i = blockIdx.x * 256 + threadIdx.x;
    if (i >= nN) return;
    float hv[CIN];
#pragma unroll
    for (int c = 0; c < CIN; ++c) hv[c] = h[(size_t)i * CIN + c];
#pragma unroll
    for (int o = 0; o < COUT; ++o) {
        float acc = init ? b[o] : out[(size_t)i * COUT + o];
        const float* wr = w + (size_t)o * wstride;
#pragma unroll
        for (int c = 0; c < CIN; ++c) acc += hv[c] * wr[c];
        out[(size_t)i * COUT + o] = acc;
    }
}

__global__ __launch_bounds__(256) void k_sigmoid(float* __restrict__ v, int n) {
    int i = blockIdx.x * 256 + threadIdx.x;
    if (i < n) v[i] = sigmoidf_(v[i]);
}

// ---------- final: out = relu([x1, relu(oa), relu(ob)] @ lw.T + lb) ----------
__global__ __launch_bounds__(256) void k_combine(const float* __restrict__ x1f, const float* __restrict__ oa,
                                                 const float* __restrict__ ob, const float* __restrict__ lw,
                                                 const float* __restrict__ lb, float* __restrict__ out, int nN) {
    int i = blockIdx.x * 256 + threadIdx.x;
    if (i < nN)
        out[i] = fmaxf(x1f[i] * lw[0] + fmaxf(oa[i], 0.f) * lw[1] + fmaxf(ob[i], 0.f) * lw[2] + lb[0], 0.0f);
}

extern "C" void kernel_launch(void* const* d_in, const int* in_sizes, int n_in,
                              void* d_out, int out_size, void* d_ws, size_t ws_size,
                              hipStream_t stream) {
    const int N = in_sizes[0] / 11;
    const int E = in_sizes[1] / 2;

    const float* x   = (const float*)d_in[0];
    const int*   ei  = (const int*)d_in[1];
    const int*   src = ei;
    const int*   dst = ei + E;
    const float* wl1 = (const float*)d_in[2];
    const float* wr1 = (const float*)d_in[3];
    const float* b1  = (const float*)d_in[4];
    const float* wl2 = (const float*)d_in[5];
    const float* wr2 = (const float*)d_in[6];
    const float* b2  = (const float*)d_in[7];
    const float* wl3 = (const float*)d_in[8];
    const float* wr3 = (const float*)d_in[9];
    const float* b3  = (const float*)d_in[10];
    const float* w1a = (const float*)d_in[11];
    const float* b1a = (const float*)d_in[12];
    const float* w2a = (const float*)d_in[13];
    const float* b2a = (const float*)d_in[14];
    const float* w1b = (const float*)d_in[15];
    const float* b1b = (const float*)d_in[16];
    const float* w2b = (const float*)d_in[17];
    const float* b2b = (const float*)d_in[18];
    const float* lw  = (const float*)d_in[19];
    const float* lb  = (const float*)d_in[20];
    float* out = (float*)d_out;

    // workspace carve (float units, 256B aligned)
    float* W = (float*)d_ws;
    size_t off = 0;
    auto alloc = [&](size_t nflt) { float* p = W + off; off += (nflt + 63) & ~(size_t)63; return p; };
    float* DEG  = alloc(N);
    float* INVC = alloc(N);
    float* DIS  = alloc(N);
    float* NRM  = alloc(E);
    float* B1   = alloc((size_t)N * 128);  // x1a (sage1 out)
    float* B2   = alloc((size_t)N * 128);  // agg128 / x1b (in-place)
    float* Y    = alloc(N);
    float* YR   = alloc(N);
    float* AGY  = alloc(N);
    float* X1F  = alloc(N);
    float* H11A = alloc((size_t)N * 11);   // also sage1's agg11
    float* H11B = alloc((size_t)N * 11);
    float* O8   = alloc((size_t)N * 8);
    float* H8A  = alloc((size_t)N * 8);
    float* H8B  = alloc((size_t)N * 8);
    float* OA   = alloc(N);
    float* OB   = alloc(N);
    (void)ws_size; (void)n_in; (void)out_size;

    const int GE = DIV_UP(E, 256), GN = DIV_UP(N, 256);

    // degree / norm
    (void)hipMemsetAsync(DEG, 0, (size_t)N * 4, stream);
    k_deg<<<GE, 256, 0, stream>>>(dst, DEG, E);
    k_dn<<<GN, 256, 0, stream>>>(DEG, INVC, DIS, N);
    k_norm<<<GE, 256, 0, stream>>>(src, dst, DIS, NRM, E);

    // ---- SAGE1 (11 -> 128) ----
    (void)hipMemsetAsync(H11A, 0, (size_t)N * 11 * 4, stream);
    k_scatter<11><<<GE, 256, 0, stream>>>(src, dst, nullptr, x, H11A, E);
    k_sage1<<<DIV_UP(N * 128, 256), 256, 0, stream>>>(H11A, x, INVC, wl1, wr1, b1, B1, N);

    // ---- SAGE2 (128 -> 128, WMMA) ----
    (void)hipMemsetAsync(B2, 0, (size_t)N * 128 * 4, stream);
    k_scatter128<<<DIV_UP(E, 8), 256, 0, stream>>>(src, dst, B1, B2, E);
    k_sage2_wmma<<<DIV_UP(N, 16), 256, 0, stream>>>(B2, B1, INVC, wl2, wr2, b2, B2, N);

    // ---- SAGE3 (128 -> 1, project-then-aggregate) ----
    k_sage3_proj<<<DIV_UP(N, 8), 256, 0, stream>>>(B2, wl3, wr3, Y, YR, N);
    (void)hipMemsetAsync(AGY, 0, (size_t)N * 4, stream);
    k_scatter<1><<<GE, 256, 0, stream>>>(src, dst, nullptr, Y, AGY, E);
    k_sage3_fin<<<GN, 256, 0, stream>>>(AGY, INVC, YR, b3, X1F, N);

    // ---- TAG1a: K=3, 11 -> 8 (w [8,44]) ----
    k_tagacc<11, 8><<<GN, 256, 0, stream>>>(x, w1a, 44, b1a, O8, N, 1);
    {
        const float* hp = x;
        float* pp[2] = {H11B, H11A};
        for (int k = 1; k <= 3; ++k) {
            float* hc = pp[k & 1];
            (void)hipMemsetAsync(hc, 0, (size_t)N * 11 * 4, stream);
            k_scatter<11><<<GE, 256, 0, stream>>>(src, dst, NRM, hp, hc, E);
            k_tagacc<11, 8><<<GN, 256, 0, stream>>>(hc, w1a + 11 * k, 44, b1a, O8, N, 0);
            hp = hc;
        }
    }
    k_sigmoid<<<DIV_UP(N * 8, 256), 256, 0, stream>>>(O8, N * 8);

    // ---- TAG2a: K=3, 8 -> 1 (w [1,32]) ----
    k_tagacc<8, 1><<<GN, 256, 0, stream>>>(O8, w2a, 32, b2a, OA, N, 1);
    {
        const float* hp = O8;
        float* pp[2] = {H8B, H8A};
        for (int k = 1; k <= 3; ++k) {
            float* hc = pp[k & 1];
            (void)hipMemsetAsync(hc, 0, (size_t)N * 8 * 4, stream);
            k_scatter<8><<<GE, 256, 0, stream>>>(src, dst, NRM, hp, hc, E);
            k_tagacc<8, 1><<<GN, 256, 0, stream>>>(hc, w2a + 8 * k, 32, b2a, OA, N, 0);
            hp = hc;
        }
    }

    // ---- TAG1b: K=6, 11 -> 8 (w [8,77]) ----
    k_tagacc<11, 8><<<GN, 256, 0, stream>>>(x, w1b, 77, b1b, O8, N, 1);
    {
        const float* hp = x;
        float* pp[2] = {H11B, H11A};
        for (int k = 1; k <= 6; ++k) {
            float* hc = pp[k & 1];
            (void)hipMemsetAsync(hc, 0, (size_t)N * 11 * 4, stream);
            k_scatter<11><<<GE, 256, 0, stream>>>(src, dst, NRM, hp, hc, E);
            k_tagacc<11, 8><<<GN, 256, 0, stream>>>(hc, w1b + 11 * k, 77, b1b, O8, N, 0);
            hp = hc;
        }
    }
    k_sigmoid<<<DIV_UP(N * 8, 256), 256, 0, stream>>>(O8, N * 8);

    // ---- TAG2b: K=6, 8 -> 1 (w [1,56]) ----
    k_tagacc<8, 1><<<GN, 256, 0, stream>>>(O8, w2b, 56, b2b, OB, N, 1);
    {
        const float* hp = O8;
        float* pp[2] = {H8B, H8A};
        for (int k = 1; k <= 6; ++k) {
            float* hc = pp[k & 1];
            (void)hipMemsetAsync(hc, 0, (size_t)N * 8 * 4, stream);
            k_scatter<8><<<GE, 256, 0, stream>>>(src, dst, NRM, hp, hc, E);
            k_tagacc<8, 1><<<GN, 256, 0, stream>>>(hc, w2b + 8 * k, 56, b2b, OB, N, 0);
            hp = hc;
        }
    }

    // ---- final combine ----
    k_combine<<<GN, 256, 0, stream>>>(X1F, OA, OB, lw, lb, out, N);
}